// ReaxBonded_57062935495398
// MI455X (gfx1250) — compile-verified
//
#include <hip/hip_runtime.h>
#include <math.h>

#define NSP 119
#define NPAIRS 6400000
#define MAIN_BLOCKS 1024
#define MAIN_THREADS 256
#define PACK_FLOATS (NSP * NSP * 12)   // 169932 floats, 48B per species pair

// Direct hardware exp2 (v_exp_f32)
#define EXP2F(x) __builtin_amdgcn_exp2f(x)

// Numerically stable softplus (matches jax.nn.softplus)
__device__ __forceinline__ float sp_stable(float x) {
    return fmaxf(x, 0.0f) + log1pf(__expf(-fabsf(x)));
}

// ---------------------------------------------------------------------------
// Kernel 1: hoist all softplus() of species-pair parameters out of the pair
// loop. Packs [pc0 pc1 pc2 pe0 | pe1 pe2 De p1 | p2 0 0 0] per (Zi,Zj) so the
// pair kernel does two aligned 16B loads + one scalar instead of 5 gathers
// and 8 softplus evaluations.
// ---------------------------------------------------------------------------
__global__ void reax_precompute(const float* __restrict__ r0,
                                const float* __restrict__ po_coeff,
                                const float* __restrict__ po_exp,
                                const float* __restrict__ De,
                                const float* __restrict__ pbe1,
                                const float* __restrict__ pbe2,
                                float* __restrict__ pack,
                                float* __restrict__ spr0) {
    int t = blockIdx.x * blockDim.x + threadIdx.x;
    if (t < NSP) spr0[t] = sp_stable(r0[t]);
    if (t >= NSP * NSP) return;
    float* o = pack + t * 12;
    o[0]  = sp_stable(po_coeff[3 * t + 0]);
    o[1]  = sp_stable(po_coeff[3 * t + 1]);
    o[2]  = sp_stable(po_coeff[3 * t + 2]);
    o[3]  = sp_stable(po_exp[3 * t + 0]);
    o[4]  = sp_stable(po_exp[3 * t + 1]);
    o[5]  = sp_stable(po_exp[3 * t + 2]);
    o[6]  = sp_stable(De[t]);
    o[7]  = pbe1[t];
    o[8]  = sp_stable(pbe2[t] + 1.0f);
    o[9]  = 0.0f; o[10] = 0.0f; o[11] = 0.0f;
}

// ---------------------------------------------------------------------------
// Kernel 2: main pair kernel. Per-wave double-buffered async copy of the
// index stream into LDS (ASYNCcnt pipeline), L2 prefetch ahead, f32
// transcendental pipeline, deterministic wave32 + LDS reduction to partials.
// ---------------------------------------------------------------------------
__global__ __launch_bounds__(MAIN_THREADS)
void reax_pair(const float* __restrict__ R,
               const int* __restrict__ Z,
               const int* __restrict__ idx,
               const float* __restrict__ pack,
               const float* __restrict__ spr0,
               float* __restrict__ partials) {
    __shared__ int stage[MAIN_THREADS / 32][2][2][32];  // [wave][buf][i/j][lane]
    __shared__ float wsum[MAIN_THREADS / 32];

    const int lane = threadIdx.x & 31;
    const int wv   = threadIdx.x >> 5;
    const int stride   = MAIN_BLOCKS * MAIN_THREADS;
    const int wavebase = blockIdx.x * MAIN_THREADS + (threadIdx.x & ~31);
    const int niter    = (NPAIRS - wavebase + stride - 1) / stride;  // uniform per wave

    // LDS byte addresses for this lane's slots (low 32 bits of flat shared ptr)
    const unsigned lds_i = (unsigned)(uintptr_t)(&stage[wv][0][0][lane]);
    const unsigned lds_j = (unsigned)(uintptr_t)(&stage[wv][0][1][lane]);
    const unsigned bufBytes = 2u * 32u * 4u;  // 256 B between buf 0 and buf 1

    // Prologue: async-stage tile 0 into buffer 0
    {
        int p = wavebase + lane;
        if (p >= NPAIRS) p = NPAIRS - 1;
        unsigned oi = (unsigned)p * 4u;
        unsigned oj = oi + (unsigned)NPAIRS * 4u;
        asm volatile("global_load_async_to_lds_b32 %0, %1, %2"
                     :: "v"(lds_i), "v"(oi), "s"(idx) : "memory");
        asm volatile("global_load_async_to_lds_b32 %0, %1, %2"
                     :: "v"(lds_j), "v"(oj), "s"(idx) : "memory");
    }

    float acc = 0.0f;
    for (int k = 0; k < niter; ++k) {
        // Issue next tile into the other buffer (clamped dummy on last iter)
        {
            int kn = (k + 1 < niter) ? (k + 1) : k;
            int p  = wavebase + lane + kn * stride;
            if (p >= NPAIRS) p = NPAIRS - 1;
            unsigned b  = (unsigned)((k + 1) & 1) * bufBytes;
            unsigned oi = (unsigned)p * 4u;
            unsigned oj = oi + (unsigned)NPAIRS * 4u;
            asm volatile("global_load_async_to_lds_b32 %0, %1, %2"
                         :: "v"(lds_i + b), "v"(oi), "s"(idx) : "memory");
            asm volatile("global_load_async_to_lds_b32 %0, %1, %2"
                         :: "v"(lds_j + b), "v"(oj), "s"(idx) : "memory");
        }
        // Prefetch index lines ~6 tiles ahead into L2 (speculative, branchless)
        {
            int pf = wavebase + lane + (k + 6) * stride;
            pf = (pf < NPAIRS) ? pf : 0;
            __builtin_prefetch(idx + pf, 0, 0);
            __builtin_prefetch(idx + NPAIRS + pf, 0, 0);
        }
        // Wait until the current buffer's 2 loads are done (2 newest may remain)
        asm volatile("s_wait_asynccnt 2" ::: "memory");

        const int i = stage[wv][k & 1][0][lane];
        const int j = stage[wv][k & 1][1][lane];
        const int p = wavebase + lane + k * stride;

        // Free-space displacement (gathers hit L2: R table is 1.2 MB)
        float dx = R[3 * j + 0] - R[3 * i + 0];
        float dy = R[3 * j + 1] - R[3 * i + 1];
        float dz = R[3 * j + 2] - R[3 * i + 2];
        float d2 = dx * dx + dy * dy + dz * dz;
        float dr = sqrtf(d2);                                   // sqrt(0)=0 == safe path
        float drc = fminf(dr, 6.0f);
        float cutoff = 0.5f * (__cosf(0.52359877559829887f * drc) + 1.0f);  // pi/R_MAX

        int zi = Z[i], zj = Z[j];
        int tt = zi * NSP + zj;
        const float4* pk4 = (const float4*)(pack + 12 * tt);
        float4 q0 = pk4[0];                 // pc0 pc1 pc2 pe0
        float4 q1 = pk4[1];                 // pe1 pe2 De  p1
        float p2v = pack[12 * tt + 8];      // p2

        float r0ij  = 0.5f * (spr0[zi] + spr0[zj]);
        float ratio = dr / r0ij;
        // safe_pow for free: log2(0)=-inf, pe>0 => exp2(-inf)=0 => exp(0)=1
        float lr = __log2f(ratio);
        float s0 = __expf(-q0.x * EXP2F(q0.w * lr));
        float s1 = __expf(-q0.y * EXP2F(q1.x * lr));
        float s2 = __expf(-q0.z * EXP2F(q1.y * lr));
        float bo = (s0 + s1 + s2) * cutoff;
        float powbo = EXP2F(p2v * __log2f(bo));                 // bo==0 -> 0
        float E = -q1.z * bo * __expf(q1.w * (1.0f - powbo));

        bool valid = (p < NPAIRS) && (i != j);                  // padding + self-pair mask
        acc += valid ? E : 0.0f;
    }

    // Deterministic wave32 tree reduce, then LDS block reduce
    for (int off = 16; off > 0; off >>= 1) acc += __shfl_down(acc, off, 32);
    if (lane == 0) wsum[wv] = acc;
    __syncthreads();
    if (threadIdx.x == 0) {
        float s = 0.0f;
        for (int w = 0; w < MAIN_THREADS / 32; ++w) s += wsum[w];
        partials[blockIdx.x] = s;
    }
}

// ---------------------------------------------------------------------------
// Kernel 3: fixed-order final reduction (no atomics -> bitwise deterministic,
// and d_out never needs pre-zeroing).
// ---------------------------------------------------------------------------
__global__ void reax_reduce(const float* __restrict__ partials,
                            float* __restrict__ out) {
    float a = 0.0f;
    for (int i = threadIdx.x; i < MAIN_BLOCKS; i += blockDim.x) a += partials[i];
    for (int off = 16; off > 0; off >>= 1) a += __shfl_down(a, off, 32);
    __shared__ float ws[8];
    int lane = threadIdx.x & 31, wv = threadIdx.x >> 5;
    if (lane == 0) ws[wv] = a;
    __syncthreads();
    if (threadIdx.x == 0) {
        float s = 0.0f;
        for (int w = 0; w < 8; ++w) s += ws[w];
        out[0] = s;
    }
}

extern "C" void kernel_launch(void* const* d_in, const int* in_sizes, int n_in,
                              void* d_out, int out_size, void* d_ws, size_t ws_size,
                              hipStream_t stream) {
    (void)in_sizes; (void)n_in; (void)out_size; (void)ws_size;
    const float* R   = (const float*)d_in[0];
    const int*   Z   = (const int*)d_in[1];
    const int*   idx = (const int*)d_in[2];
    const float* r0  = (const float*)d_in[3];
    const float* pc  = (const float*)d_in[4];
    const float* pe  = (const float*)d_in[5];
    const float* De  = (const float*)d_in[6];
    const float* p1  = (const float*)d_in[7];
    const float* p2  = (const float*)d_in[8];

    float* pack     = (float*)d_ws;              // 169932 floats (48B/pair, 16B aligned)
    float* spr0     = pack + PACK_FLOATS;        // 119 floats
    float* partials = pack + PACK_FLOATS + 132;  // MAIN_BLOCKS floats (16B aligned)

    reax_precompute<<<(NSP * NSP + 255) / 256, 256, 0, stream>>>(
        r0, pc, pe, De, p1, p2, pack, spr0);
    reax_pair<<<MAIN_BLOCKS, MAIN_THREADS, 0, stream>>>(
        R, Z, idx, pack, spr0, partials);
    reax_reduce<<<1, 256, 0, stream>>>(partials, (float*)d_out);
}